// MACR_21852793602109
// MI455X (gfx1250) — compile-verified
//
#include <hip/hip_runtime.h>
#include <math.h>

// ---------------- problem constants (match reference) ----------------
#define EMB    64
#define ALPHA  1e-3f
#define BETA   1e-3f
#define EPSC   1e-10f

typedef __attribute__((ext_vector_type(2))) float v2f;
typedef __attribute__((ext_vector_type(8))) float v8f;

__device__ __forceinline__ float sigm(float x) { return 1.0f / (1.0f + expf(-x)); }

// D = A(16x4,f32) * B(4x16,f32) + C(16x16,f32), wave32, VOP3P
__device__ __forceinline__ v8f wmma4_f32(v2f a, v2f b, v8f c) {
    return __builtin_amdgcn_wmma_f32_16x16x4_f32(
        /*neg_a=*/false, a, /*neg_b=*/false, b,
        /*c_mod=*/(short)0, c, /*reuse_a=*/false, /*reuse_b=*/false);
}

// ---------------- utility kernels ----------------
__global__ void k_zero(float* __restrict__ p, size_t n) {
    size_t i = (size_t)blockIdx.x * blockDim.x + threadIdx.x;
    if (i < n) p[i] = 0.0f;
}

__global__ void k_degree(const int* __restrict__ eu, const int* __restrict__ ei,
                         float* __restrict__ du, float* __restrict__ di, int E) {
    int e = blockIdx.x * blockDim.x + threadIdx.x;
    if (e < E) {
        atomicAdd(du + eu[e], 1.0f);
        atomicAdd(di + ei[e], 1.0f);
    }
}

// deg -> clip(deg,1)^-0.5 in place
__global__ void k_norm(float* __restrict__ d, int n) {
    int i = blockIdx.x * blockDim.x + threadIdx.x;
    if (i < n) d[i] = 1.0f / sqrtf(fmaxf(d[i], 1.0f));
}

// One wave32 slice (32 lanes, 2 floats/lane) per edge; both propagation
// directions in a single pass over the edge list (reads "cur", writes "next").
// msg_i2u = hi[it]*nrm_i[it] scattered to hu_next[u]; msg_u2i symmetric.
__global__ __launch_bounds__(256)
void k_edge_scatter(const int* __restrict__ eu, const int* __restrict__ ei,
                    const float* __restrict__ hu, const float* __restrict__ hi,
                    const float* __restrict__ nu, const float* __restrict__ ni,
                    float* __restrict__ hun, float* __restrict__ hin, int E) {
    int lane = threadIdx.x & 31;
    int e = blockIdx.x * 8 + (threadIdx.x >> 5);
    if (e >= E) return;
    int u = eu[e], it = ei[e];
    int c = lane * 2;
    size_t ub = (size_t)u * EMB + c;
    size_t ib = (size_t)it * EMB + c;
    float2 vi = *(const float2*)(hi + ib);
    float2 vu = *(const float2*)(hu + ub);
    float si = ni[it], su = nu[u];
    atomicAdd(hun + ub,     vi.x * si);
    atomicAdd(hun + ub + 1, vi.y * si);
    atomicAdd(hin + ib,     vu.x * su);
    atomicAdd(hin + ib + 1, vu.y * su);
    // CDNA5 speculative prefetch of a future edge's feature rows into cache
    int ep = e + 4096;
    if (ep < E) {
        __builtin_prefetch(hi + (size_t)ei[ep] * EMB + c, 0, 1);
        __builtin_prefetch(hu + (size_t)eu[ep] * EMB + c, 0, 1);
    }
}

// post-scatter symmetric scale: v[row][*] *= nrm[row] (in place)
__global__ void k_scale(const float* __restrict__ nrm, float* __restrict__ v, size_t n) {
    size_t i = (size_t)blockIdx.x * blockDim.x + threadIdx.x;
    if (i < n) v[i] *= nrm[i >> 6];
}

// accumulate the batch-gathered rows of this layer into small [B,EMB] buffers
__global__ void k_gather_acc(const float* __restrict__ hu, const float* __restrict__ hi,
                             const int* __restrict__ user, const int* __restrict__ itp,
                             const int* __restrict__ itn,
                             float* __restrict__ ua, float* __restrict__ pa,
                             float* __restrict__ na, int B) {
    int idx = blockIdx.x * blockDim.x + threadIdx.x;
    if (idx < B * EMB) {
        int b = idx >> 6, c = idx & 63;
        ua[idx] += hu[(size_t)user[b] * EMB + c];
        pa[idx] += hi[(size_t)itp[b]  * EMB + c];
        na[idx] += hi[(size_t)itn[b]  * EMB + c];
    }
}

// ---------------- WMMA batch scoring ----------------
// One wave (32 threads) per 16 batch rows. Five f32 WMMA chains over K=64:
//   pos = diag(uf @ pf^T)/64, neg = diag(uf @ nf^T)/64,
//   pis = pf@Wi + bi, nis = nf@Wi + bi, usc = uf@Wu + bu.
// A(16x4) layout: lane l -> M=l&15, VGPR0/1 hold K = k0+2*(l>>4) .. +1.
// B(4x16) layout mirrors A with M<->N, so B for P^T uses the SAME per-lane
// values as A for P. Weight-column B is nonzero only in lanes 0 and 16.
// D(16x16): VGPR r, lanes0-15 -> (M=r, N=lane); lanes16-31 -> (M=8+r, N=lane-16).
__global__ __launch_bounds__(32)
void k_scores_wmma(const float* __restrict__ ua, const float* __restrict__ pa,
                   const float* __restrict__ na,
                   const float* __restrict__ Wu, const float* __restrict__ Wi,
                   const float* __restrict__ bu, const float* __restrict__ bi,
                   float* __restrict__ pos, float* __restrict__ neg,
                   float* __restrict__ pis, float* __restrict__ nis,
                   float* __restrict__ usc) {
    __shared__ float U[16][65], P[16][65], Nf[16][65];  // padded stride: no bank conflicts
    const int l = threadIdx.x;
    const int base = blockIdx.x * 16;

    // stage tiles (mean over 4 layer embeddings = accumulated sum * 0.25)
    for (int idx = l; idx < 16 * EMB; idx += 32) {
        int m = idx >> 6, c = idx & 63;
        size_t g = (size_t)(base + m) * EMB + c;
        U[m][c]  = ua[g] * 0.25f;
        P[m][c]  = pa[g] * 0.25f;
        Nf[m][c] = na[g] * 0.25f;
    }
    __syncthreads();

    const int mn = l & 15;
    const int kh = (l >> 4) * 2;
    v8f dpos = {}, dneg = {}, dpis = {}, dnis = {}, dusc = {};

    for (int k0 = 0; k0 < EMB; k0 += 4) {
        int ka = k0 + kh;
        v2f aU = { U[mn][ka],  U[mn][ka + 1]  };   // A operand of uf  (== B of uf^T)
        v2f aP = { P[mn][ka],  P[mn][ka + 1]  };   // A of pf (== B of pf^T)
        v2f aN = { Nf[mn][ka], Nf[mn][ka + 1] };   // A of nf (== B of nf^T)
        v2f bWi = { 0.0f, 0.0f }, bWu = { 0.0f, 0.0f };
        if (mn == 0) {            // column 0 of the padded 4x16 weight B-matrix
            bWi = (v2f){ Wi[ka], Wi[ka + 1] };
            bWu = (v2f){ Wu[ka], Wu[ka + 1] };
        }
        dpos = wmma4_f32(aU, aP,  dpos);
        dneg = wmma4_f32(aU, aN,  dneg);
        dpis = wmma4_f32(aP, bWi, dpis);
        dnis = wmma4_f32(aN, bWi, dnis);
        dusc = wmma4_f32(aU, bWu, dusc);
    }

    const float bif = bi[0], buf_ = bu[0];
    if (l == 0) {
#pragma unroll
        for (int r = 0; r < 8; ++r) {
            pis[base + r] = dpis[r] + bif;
            nis[base + r] = dnis[r] + bif;
            usc[base + r] = dusc[r] + buf_;
        }
    } else if (l == 16) {
#pragma unroll
        for (int r = 0; r < 8; ++r) {
            pis[base + 8 + r] = dpis[r] + bif;
            nis[base + 8 + r] = dnis[r] + bif;
            usc[base + 8 + r] = dusc[r] + buf_;
        }
    }
    // diagonal extraction: m<8 -> lane m, vgpr m ; m>=8 -> lane m+16, vgpr m-8
    int ridx = (l < 8) ? l : (l >= 24 ? l - 24 : -1);
    float vp = 0.0f, vn = 0.0f;
#pragma unroll
    for (int r = 0; r < 8; ++r)
        if (r == ridx) { vp = dpos[r]; vn = dneg[r]; }
    if (l < 8) {
        pos[base + l] = vp * (1.0f / 64.0f);
        neg[base + l] = vn * (1.0f / 64.0f);
    } else if (l >= 24) {
        pos[base + l - 16] = vp * (1.0f / 64.0f);
        neg[base + l - 16] = vn * (1.0f / 64.0f);
    }
}

// ---------------- loss kernels ----------------
// a[i]=sig(pis)*sig(usc), b[i]=sig(nis)*sig(usc); also S_item, S_user partials
__global__ __launch_bounds__(256)
void k_batch_small(const float* __restrict__ pis, const float* __restrict__ nis,
                   const float* __restrict__ usc,
                   float* __restrict__ aa, float* __restrict__ bb,
                   float* __restrict__ part, int B) {
    __shared__ float r0[256], r1[256];
    int i = blockIdx.x * blockDim.x + threadIdx.x;
    float ti = 0.0f, tu = 0.0f;
    if (i < B) {
        float sp = sigm(pis[i]), sn = sigm(nis[i]), su = sigm(usc[i]);
        aa[i] = sp * su;
        bb[i] = sn * su;
        ti = logf(sp + EPSC) + logf(1.0f - sn + EPSC);
        tu = logf(su + EPSC) + logf(1.0f - su + EPSC);
    }
    r0[threadIdx.x] = ti; r1[threadIdx.x] = tu;
    __syncthreads();
    for (int s = 128; s > 0; s >>= 1) {
        if ((int)threadIdx.x < s) { r0[threadIdx.x] += r0[threadIdx.x + s];
                                    r1[threadIdx.x] += r1[threadIdx.x + s]; }
        __syncthreads();
    }
    if (threadIdx.x == 0) { atomicAdd(part + 1, r0[0]); atomicAdd(part + 2, r1[0]); }
}

// S_ori over the full [B,B] broadcast: pos_s[i,j] = pos[j]*a[i], etc.
__global__ __launch_bounds__(256)
void k_bbloss(const float* __restrict__ pos, const float* __restrict__ neg,
              const float* __restrict__ aa, const float* __restrict__ bb,
              float* __restrict__ part, int B) {
    __shared__ float red[256];
    size_t total  = (size_t)B * (size_t)B;
    size_t stride = (size_t)gridDim.x * blockDim.x;
    float local = 0.0f;
    for (size_t idx = (size_t)blockIdx.x * blockDim.x + threadIdx.x; idx < total; idx += stride) {
        int i = (int)(idx / (size_t)B);
        int j = (int)(idx - (size_t)i * B);
        float sp = sigm(pos[j] * aa[i]);
        float sn = sigm(neg[j] * bb[i]);
        local += logf(sp + EPSC) + logf(1.0f - sn + EPSC);
    }
    red[threadIdx.x] = local;
    __syncthreads();
    for (int s = 128; s > 0; s >>= 1) {
        if ((int)threadIdx.x < s) red[threadIdx.x] += red[threadIdx.x + s];
        __syncthreads();
    }
    if (threadIdx.x == 0) atomicAdd(part + 0, red[0]);
}

__global__ void k_finalize(const float* __restrict__ part, float* __restrict__ out, int B) {
    float Bf = (float)B;
    out[0] = -part[0] / (Bf * Bf) - ALPHA * part[1] / Bf - BETA * part[2] / Bf;
}

// ---------------- host launcher ----------------
extern "C" void kernel_launch(void* const* d_in, const int* in_sizes, int n_in,
                              void* d_out, int out_size, void* d_ws, size_t ws_size,
                              hipStream_t stream) {
    const float* emb_user = (const float*)d_in[0];
    const float* emb_item = (const float*)d_in[1];
    const float* Wu = (const float*)d_in[2];
    const float* bu = (const float*)d_in[3];
    const float* Wi = (const float*)d_in[4];
    const float* bi = (const float*)d_in[5];
    const int* user   = (const int*)d_in[6];
    const int* item_p = (const int*)d_in[7];
    const int* item_n = (const int*)d_in[8];
    const int* edge_u = (const int*)d_in[9];
    const int* edge_i = (const int*)d_in[10];

    const int N_U = in_sizes[0] / EMB;      // 200000
    const int N_I = in_sizes[1] / EMB;      // 100000
    const int B   = in_sizes[6];            // 4096
    const int E   = in_sizes[9];            // 2000000

    // workspace layout (floats) — total ~158 MB
    float* ws = (float*)d_ws;
    size_t o = 0;
    float* nrm_u = ws + o; o += (size_t)N_U;
    float* nrm_i = ws + o; o += (size_t)N_I;
    float* hu_a  = ws + o; o += (size_t)N_U * EMB;
    float* hu_b  = ws + o; o += (size_t)N_U * EMB;
    float* hi_a  = ws + o; o += (size_t)N_I * EMB;
    float* hi_b  = ws + o; o += (size_t)N_I * EMB;
    float* uf    = ws + o; o += (size_t)B * EMB;   // sum of 4 layer embeddings (batch rows)
    float* pf    = ws + o; o += (size_t)B * EMB;
    float* nf    = ws + o; o += (size_t)B * EMB;
    float* pos   = ws + o; o += (size_t)B;
    float* neg   = ws + o; o += (size_t)B;
    float* pis   = ws + o; o += (size_t)B;
    float* nis   = ws + o; o += (size_t)B;
    float* usc   = ws + o; o += (size_t)B;
    float* aa    = ws + o; o += (size_t)B;
    float* bb    = ws + o; o += (size_t)B;
    float* part  = ws + o; o += 8;

    const unsigned T = 256;
    auto blocks = [](size_t n, unsigned t) { return (unsigned)((n + t - 1) / t); };

    // 1) degrees -> symmetric norms
    k_zero<<<blocks((size_t)N_U + N_I, T), T, 0, stream>>>(nrm_u, (size_t)N_U + N_I);
    k_degree<<<blocks((size_t)E, T), T, 0, stream>>>(edge_u, edge_i, nrm_u, nrm_i, E);
    k_norm<<<blocks((size_t)N_U, T), T, 0, stream>>>(nrm_u, N_U);
    k_norm<<<blocks((size_t)N_I, T), T, 0, stream>>>(nrm_i, N_I);

    // 2) layer-0 batch accumulation (us[0] = emb)
    k_zero<<<blocks((size_t)3 * B * EMB, T), T, 0, stream>>>(uf, (size_t)3 * B * EMB);
    k_gather_acc<<<blocks((size_t)B * EMB, T), T, 0, stream>>>(
        emb_user, emb_item, user, item_p, item_n, uf, pf, nf, B);

    // 3) 3 propagation layers (ping-pong; layer 1 reads the input embeddings)
    const float* cur_u = emb_user;
    const float* cur_i = emb_item;
    float* Ubuf[2] = { hu_a, hu_b };
    float* Ibuf[2] = { hi_a, hi_b };
    for (int L = 0; L < 3; ++L) {
        float* nxt_u = Ubuf[L & 1];
        float* nxt_i = Ibuf[L & 1];
        k_zero<<<blocks((size_t)N_U * EMB, T), T, 0, stream>>>(nxt_u, (size_t)N_U * EMB);
        k_zero<<<blocks((size_t)N_I * EMB, T), T, 0, stream>>>(nxt_i, (size_t)N_I * EMB);
        k_edge_scatter<<<(unsigned)((E + 7) / 8), T, 0, stream>>>(
            edge_u, edge_i, cur_u, cur_i, nrm_u, nrm_i, nxt_u, nxt_i, E);
        k_scale<<<blocks((size_t)N_U * EMB, T), T, 0, stream>>>(nrm_u, nxt_u, (size_t)N_U * EMB);
        k_scale<<<blocks((size_t)N_I * EMB, T), T, 0, stream>>>(nrm_i, nxt_i, (size_t)N_I * EMB);
        k_gather_acc<<<blocks((size_t)B * EMB, T), T, 0, stream>>>(
            nxt_u, nxt_i, user, item_p, item_n, uf, pf, nf, B);
        cur_u = nxt_u;
        cur_i = nxt_i;
    }

    // 4) WMMA batch scoring: 256 tiles of 16 rows, one wave each
    k_scores_wmma<<<(unsigned)(B / 16), 32, 0, stream>>>(
        uf, pf, nf, Wu, Wi, bu, bi, pos, neg, pis, nis, usc);

    // 5) loss reductions
    k_zero<<<1, 8, 0, stream>>>(part, 8);
    k_batch_small<<<blocks((size_t)B, T), T, 0, stream>>>(pis, nis, usc, aa, bb, part, B);
    k_bbloss<<<2048, T, 0, stream>>>(pos, neg, aa, bb, part, B);
    k_finalize<<<1, 1, 0, stream>>>(part, (float*)d_out, B);
}